// DHPM_38946763440303
// MI455X (gfx1250) — compile-verified
//
#include <hip/hip_runtime.h>

typedef float v2f __attribute__((ext_vector_type(2)));
typedef float v8f __attribute__((ext_vector_type(8)));

#define HID 200
#define HP  208     // padded neuron dim (13 tiles of 16)
#define HPL 212     // LDS row stride in floats (bank spreading)
#define KP  100     // k-pairs
#define NTL 13      // N-tiles
#define THREADS 128

__device__ __constant__ int c_umap[6] = {0, 4, 6, 2, 5, 7};    // u, u_x, u_y, u_t, u_xx, u_yy -> out cols
__device__ __constant__ int c_vmap[6] = {1, 8, 10, 3, 9, 11};  // v, v_x, v_y, v_t, v_xx, v_yy -> out cols

static __device__ __forceinline__ v8f wmma_f32(v2f a, v2f b, v8f c) {
  // D(16x16,f32) = A(16x4,f32) * B(4x16,f32) + C
  return __builtin_amdgcn_wmma_f32_16x16x4_f32(false, a, false, b, (short)0, c, false, false);
}

// Pre-transpose W1 (200x200 row-major, z2 = W1*h1) into B layout:
// ws[w][kp*HP + n] = float2{ W1[n][2kp], W1[n][2kp+1] }, zero-padded for n >= 200.
__global__ __launch_bounds__(256) void dhpm_tr(const float* __restrict__ Wu1,
                                               const float* __restrict__ Wv1,
                                               const float* __restrict__ Wf1,
                                               float2* __restrict__ ws) {
  int idx = blockIdx.x * 256 + threadIdx.x;
  const int per = KP * HP;
  if (idx >= 3 * per) return;
  int w   = idx / per;
  int rem = idx - w * per;
  int kp  = rem / HP;
  int n   = rem - kp * HP;
  const float* W = (w == 0) ? Wu1 : (w == 1) ? Wv1 : Wf1;
  float2 v; v.x = 0.f; v.y = 0.f;
  if (n < HID) { v.x = W[n * HID + 2 * kp]; v.y = W[n * HID + 2 * kp + 1]; }
  ws[idx] = v;
}

__global__ __launch_bounds__(THREADS) void dhpm_main(
    const float* __restrict__ xin,
    const float* __restrict__ Wu0, const float* __restrict__ bu0,
    const float* __restrict__ bu1, const float* __restrict__ Wu2,
    const float* __restrict__ bu2,
    const float* __restrict__ Wv0, const float* __restrict__ bv0,
    const float* __restrict__ bv1, const float* __restrict__ Wv2,
    const float* __restrict__ bv2,
    const float* __restrict__ Wf0, const float* __restrict__ bf0,
    const float* __restrict__ bf1, const float* __restrict__ Wf2,
    const float* __restrict__ bf2,
    const float2* __restrict__ wsU, const float2* __restrict__ wsV,
    const float2* __restrict__ wsF,
    float* __restrict__ out, int Npts)
{
  __shared__ float sH[16][HPL];   // sin(z1) / later sin(z1f)
  __shared__ float sC[16][HPL];   // cos(z1)
  __shared__ float sS[5][HID];    // w0x, w0y, w0t, -w0x^2, -w0y^2
  __shared__ float sX[16][3];
  __shared__ float sFeat[16][12];
  __shared__ float sAcc[16][6];

  const int tid  = threadIdx.x;
  const int lane = tid & 31;
  const int wave = tid >> 5;
  const int hlf  = lane >> 4;     // which 16-lane half
  const int ln   = lane & 15;
  const int m0   = blockIdx.x * 16;

  if (tid < 48) {
    int m = tid / 3, c = tid - 3 * m;
    int gp = m0 + m;
    sX[m][c] = (gp < Npts) ? xin[(size_t)gp * 3 + c] : 0.f;
  }
  if (tid < 96) sAcc[tid / 6][tid % 6] = 0.f;
  __syncthreads();

  // ---------------- u and v networks (forward + tangents) ----------------
  for (int mlp = 0; mlp < 2; ++mlp) {
    const float*  W0  = mlp ? Wv0 : Wu0;
    const float*  b0  = mlp ? bv0 : bu0;
    const float*  b1  = mlp ? bv1 : bu1;
    const float*  W2  = mlp ? Wv2 : Wu2;
    const float*  b2  = mlp ? bv2 : bu2;
    const float2* wsB = mlp ? wsV : wsU;

    // per-neuron tangent scales from W0 columns
    for (int k = tid; k < HID; k += THREADS) {
      float wx = W0[k * 3 + 0], wy = W0[k * 3 + 1], wt = W0[k * 3 + 2];
      sS[0][k] = wx; sS[1][k] = wy; sS[2][k] = wt;
      sS[3][k] = -wx * wx; sS[4][k] = -wy * wy;
    }
    // layer 1: z1 = W0*x + b0 ; store sin/cos
    for (int idx = tid; idx < 16 * HID; idx += THREADS) {
      int m = idx / HID, k = idx - m * HID;
      float z = W0[k * 3 + 0] * sX[m][0] + W0[k * 3 + 1] * sX[m][1]
              + W0[k * 3 + 2] * sX[m][2] + b0[k];
      sH[m][k] = __sinf(z);
      sC[m][k] = __cosf(z);
    }
    __syncthreads();

    float prt[6][8];
    #pragma unroll
    for (int q = 0; q < 6; ++q)
      #pragma unroll
      for (int r = 0; r < 8; ++r) prt[q][r] = 0.f;

    for (int t = wave; t < NTL; t += 4) {
      int n = t * 16 + ln;
      float w2v  = (n < HID) ? W2[n] : 0.f;
      float bias = (n < HID) ? b1[n] : 0.f;

      v8f acc[6];
      #pragma unroll
      for (int q = 0; q < 6; ++q)
        #pragma unroll
        for (int r = 0; r < 8; ++r) acc[q][r] = (q == 0) ? bias : 0.f;

      for (int k0 = 0; k0 < HID; k0 += 4) {
        int kb = k0 + 2 * hlf;
        v2f ah = *(const v2f*)&sH[ln][kb];
        v2f ac = *(const v2f*)&sC[ln][kb];
        v2f sx = *(const v2f*)&sS[0][kb];
        v2f sy = *(const v2f*)&sS[1][kb];
        v2f st = *(const v2f*)&sS[2][kb];
        v2f qx = *(const v2f*)&sS[3][kb];
        v2f qy = *(const v2f*)&sS[4][kb];
        float2 bp = wsB[(size_t)(k0 / 2 + hlf) * HP + n];
        v2f b; b.x = bp.x; b.y = bp.y;
        acc[0] = wmma_f32(ah,      b, acc[0]);   // z2
        acc[1] = wmma_f32(ac * sx, b, acc[1]);   // dz2/dx
        acc[2] = wmma_f32(ac * sy, b, acc[2]);   // dz2/dy
        acc[3] = wmma_f32(ac * st, b, acc[3]);   // dz2/dt
        acc[4] = wmma_f32(ah * qx, b, acc[4]);   // d2z2/dx2 (pre-chain)
        acc[5] = wmma_f32(ah * qy, b, acc[5]);   // d2z2/dy2 (pre-chain)
      }
      #pragma unroll
      for (int r = 0; r < 8; ++r) {
        float z2 = acc[0][r];
        float h2 = __sinf(z2), c2 = __cosf(z2);
        float dx = acc[1][r], dy = acc[2][r], dt = acc[3][r];
        float xx = acc[4][r], yy = acc[5][r];
        prt[0][r] += w2v * h2;
        prt[1][r] += w2v * (c2 * dx);
        prt[2][r] += w2v * (c2 * dy);
        prt[3][r] += w2v * (c2 * dt);
        prt[4][r] += w2v * (c2 * xx - h2 * dx * dx);
        prt[5][r] += w2v * (c2 * yy - h2 * dy * dy);
      }
    }
    #pragma unroll
    for (int q = 0; q < 6; ++q)
      #pragma unroll
      for (int r = 0; r < 8; ++r)
        atomicAdd(&sAcc[r + 8 * hlf][q], prt[q][r]);
    __syncthreads();

    if (tid < 96) {
      int m = tid / 6, q = tid - m * 6;
      float val = sAcc[m][q] + ((q == 0) ? b2[0] : 0.f);
      int col = mlp ? c_vmap[q] : c_umap[q];
      sFeat[m][col] = val;
      int gp = m0 + m;
      if (gp < Npts) out[(size_t)col * Npts + gp] = val;
      sAcc[m][q] = 0.f;                  // reset for next stage
    }
    __syncthreads();
  }

  // ---------------- f network: 12 -> 200 -> 200 -> 2 ----------------
  for (int idx = tid; idx < 16 * HID; idx += THREADS) {
    int m = idx / HID, k = idx - m * HID;
    float z = bf0[k];
    #pragma unroll
    for (int i = 0; i < 12; ++i) z += Wf0[k * 12 + i] * sFeat[m][i];
    sH[m][k] = __sinf(z);
  }
  __syncthreads();

  float pf0[8], pf1[8];
  #pragma unroll
  for (int r = 0; r < 8; ++r) { pf0[r] = 0.f; pf1[r] = 0.f; }

  for (int t = wave; t < NTL; t += 4) {
    int n = t * 16 + ln;
    float w20  = (n < HID) ? Wf2[n] : 0.f;
    float w21  = (n < HID) ? Wf2[HID + n] : 0.f;
    float bias = (n < HID) ? bf1[n] : 0.f;
    v8f acc;
    #pragma unroll
    for (int r = 0; r < 8; ++r) acc[r] = bias;
    for (int k0 = 0; k0 < HID; k0 += 4) {
      int kb = k0 + 2 * hlf;
      v2f a = *(const v2f*)&sH[ln][kb];
      float2 bp = wsF[(size_t)(k0 / 2 + hlf) * HP + n];
      v2f b; b.x = bp.x; b.y = bp.y;
      acc = wmma_f32(a, b, acc);
    }
    #pragma unroll
    for (int r = 0; r < 8; ++r) {
      float h2 = __sinf(acc[r]);
      pf0[r] += w20 * h2;
      pf1[r] += w21 * h2;
    }
  }
  #pragma unroll
  for (int r = 0; r < 8; ++r) {
    atomicAdd(&sAcc[r + 8 * hlf][0], pf0[r]);
    atomicAdd(&sAcc[r + 8 * hlf][1], pf1[r]);
  }
  __syncthreads();
  if (tid < 32) {
    int m = tid >> 1, j = tid & 1;
    int gp = m0 + m;
    if (gp < Npts) out[(size_t)(12 + j) * Npts + gp] = sAcc[m][j] + bf2[j];
  }
}

extern "C" void kernel_launch(void* const* d_in, const int* in_sizes, int n_in,
                              void* d_out, int out_size, void* d_ws, size_t ws_size,
                              hipStream_t stream) {
  const float* xin = (const float*)d_in[0];
  const float* Wu0 = (const float*)d_in[1];
  const float* bu0 = (const float*)d_in[2];
  const float* Wu1 = (const float*)d_in[3];
  const float* bu1 = (const float*)d_in[4];
  const float* Wu2 = (const float*)d_in[5];
  const float* bu2 = (const float*)d_in[6];
  const float* Wv0 = (const float*)d_in[7];
  const float* bv0 = (const float*)d_in[8];
  const float* Wv1 = (const float*)d_in[9];
  const float* bv1 = (const float*)d_in[10];
  const float* Wv2 = (const float*)d_in[11];
  const float* bv2 = (const float*)d_in[12];
  const float* Wf0 = (const float*)d_in[13];
  const float* bf0 = (const float*)d_in[14];
  const float* Wf1 = (const float*)d_in[15];
  const float* bf1 = (const float*)d_in[16];
  const float* Wf2 = (const float*)d_in[17];
  const float* bf2 = (const float*)d_in[18];

  int Npts = in_sizes[0] / 3;
  float2* ws = (float2*)d_ws;
  const int per = KP * HP;               // float2 elements per transposed matrix
  int trN = 3 * per;

  dhpm_tr<<<(trN + 255) / 256, 256, 0, stream>>>(Wu1, Wv1, Wf1, ws);
  dhpm_main<<<(Npts + 15) / 16, THREADS, 0, stream>>>(
      xin, Wu0, bu0, bu1, Wu2, bu2,
      Wv0, bv0, bv1, Wv2, bv2,
      Wf0, bf0, bf1, Wf2, bf2,
      ws, ws + per, ws + 2 * per,
      (float*)d_out, Npts);
}